// DSEBlock_31636729102407
// MI455X (gfx1250) — compile-verified
//
#include <hip/hip_runtime.h>
#include <hip/hip_bf16.h>
#include <math.h>

// Problem constants (reference: B=16, C=256, H=W=96, fp32)
constexpr int NB = 16;
constexpr int NC = 256;
constexpr int NH = 96;
constexpr int NW = 96;
constexpr int NHW = NH * NW;            // 9216

typedef float v2f __attribute__((ext_vector_type(2)));
typedef float v8f __attribute__((ext_vector_type(8)));
typedef unsigned int u32x4 __attribute__((ext_vector_type(4)));
typedef int i32x4 __attribute__((ext_vector_type(4)));
typedef int i32x8 __attribute__((ext_vector_type(8)));

__device__ __forceinline__ int clampi(int v, int lo, int hi) {
    return v < lo ? lo : (v > hi ? hi : v);
}

// ---------------------------------------------------------------------------
// Kernel 1: per-channel DoG kernel difference: kd[c][9] = g1[c] - g2[c]
// sigma = 2*sigmoid(raw); 3x3 gaussian exp(-r2/(2 sig^2)) normalized.
// ---------------------------------------------------------------------------
__global__ void kdiff_kernel(const float* __restrict__ s1raw,
                             const float* __restrict__ s2raw,
                             float* __restrict__ kd) {
    int c = blockIdx.x * blockDim.x + threadIdx.x;
    if (c >= NC) return;
    const float r2[9] = {2.f, 1.f, 2.f, 1.f, 0.f, 1.f, 2.f, 1.f, 2.f};
    float sg1 = 2.f / (1.f + expf(-s1raw[c]));
    float sg2 = 2.f / (1.f + expf(-s2raw[c]));
    float i1 = 1.f / (2.f * sg1 * sg1);
    float i2 = 1.f / (2.f * sg2 * sg2);
    float k1[9], k2[9];
    float sum1 = 0.f, sum2 = 0.f;
#pragma unroll
    for (int t = 0; t < 9; ++t) {
        k1[t] = expf(-r2[t] * i1); sum1 += k1[t];
        k2[t] = expf(-r2[t] * i2); sum2 += k2[t];
    }
    float n1 = 1.f / sum1, n2 = 1.f / sum2;
#pragma unroll
    for (int t = 0; t < 9; ++t)
        kd[c * 9 + t] = k1[t] * n1 - k2[t] * n2;
}

// ---------------------------------------------------------------------------
// Kernel 2: bilinear (half-pixel, edge-clamped) downsample of y = dec + skip
// ---------------------------------------------------------------------------
__global__ void down_kernel(const float* __restrict__ dec,
                            const float* __restrict__ skip,
                            float* __restrict__ dst, int O) {
    int total = NB * NC * O * O;
    int idx = blockIdx.x * blockDim.x + threadIdx.x;
    if (idx >= total) return;
    int ow = idx % O;
    int oh = (idx / O) % O;
    int bc = idx / (O * O);
    float scale = (float)NH / (float)O;
    float sy = (oh + 0.5f) * scale - 0.5f;
    float sx = (ow + 0.5f) * scale - 0.5f;
    int y0 = (int)floorf(sy), x0 = (int)floorf(sx);
    float fy = sy - (float)y0, fx = sx - (float)x0;
    int y0c = clampi(y0, 0, NH - 1), y1c = clampi(y0 + 1, 0, NH - 1);
    int x0c = clampi(x0, 0, NW - 1), x1c = clampi(x0 + 1, 0, NW - 1);
    const float* db = dec  + (size_t)bc * NHW;
    const float* sb = skip + (size_t)bc * NHW;
    int o00 = y0c * NW + x0c, o01 = y0c * NW + x1c;
    int o10 = y1c * NW + x0c, o11 = y1c * NW + x1c;
    float v00 = db[o00] + sb[o00], v01 = db[o01] + sb[o01];
    float v10 = db[o10] + sb[o10], v11 = db[o11] + sb[o11];
    dst[idx] = (1.f - fy) * ((1.f - fx) * v00 + fx * v01) +
               fy         * ((1.f - fx) * v10 + fx * v11);
}

// ---------------------------------------------------------------------------
// Kernel 3: fuse pre-GEMM: z = 3*y + w_fea[c]*w_edge + dog
// ---------------------------------------------------------------------------
__global__ void fuse_kernel(const float* __restrict__ dec,
                            const float* __restrict__ skip,
                            const float* __restrict__ d1,
                            const float* __restrict__ d2,
                            const float* __restrict__ d3,
                            const float* __restrict__ kd,
                            const float* __restrict__ wfea,
                            float* __restrict__ zin) {
    int idx = blockIdx.x * blockDim.x + threadIdx.x;
    if (idx >= NB * NC * NHW) return;
    int w = idx % NW;
    int h = (idx / NW) % NH;
    int bc = idx / NHW;
    int c = bc % NC;
    const float* db = dec  + (size_t)bc * NHW;
    const float* sb = skip + (size_t)bc * NHW;
    float yc = db[h * NW + w] + sb[h * NW + w];

    // DoG: depthwise 3x3, zero padding (SAME), kernel = g1-g2
    float dog = 0.f;
    const float* kdc = kd + c * 9;
#pragma unroll
    for (int dy = -1; dy <= 1; ++dy) {
#pragma unroll
        for (int dx = -1; dx <= 1; ++dx) {
            int hh = h + dy, ww = w + dx;
            if (hh >= 0 && hh < NH && ww >= 0 && ww < NW) {
                int o = hh * NW + ww;
                dog += kdc[(dy + 1) * 3 + (dx + 1)] * (db[o] + sb[o]);
            }
        }
    }

    // Multi-scale edges: |y - up(down(y))| at 24/48/72
    const float* ds[3] = {d1, d2, d3};
    const int osz[3] = {24, 48, 72};
    float e[3];
#pragma unroll
    for (int s = 0; s < 3; ++s) {
        int O = osz[s];
        float scale = (float)O / (float)NH;
        float sy = (h + 0.5f) * scale - 0.5f;
        float sx = (w + 0.5f) * scale - 0.5f;
        int y0 = (int)floorf(sy), x0 = (int)floorf(sx);
        float fy = sy - (float)y0, fx = sx - (float)x0;
        int y0c = clampi(y0, 0, O - 1), y1c = clampi(y0 + 1, 0, O - 1);
        int x0c = clampi(x0, 0, O - 1), x1c = clampi(x0 + 1, 0, O - 1);
        const float* p = ds[s] + (size_t)bc * O * O;
        float v00 = p[y0c * O + x0c], v01 = p[y0c * O + x1c];
        float v10 = p[y1c * O + x0c], v11 = p[y1c * O + x1c];
        float up = (1.f - fy) * ((1.f - fx) * v00 + fx * v01) +
                   fy         * ((1.f - fx) * v10 + fx * v11);
        e[s] = fabsf(yc - up);
    }
    float we = (fabsf(e[0] - e[1]) + fabsf(e[0] - e[2]) + fabsf(e[1] - e[2])) * (1.f / 3.f);
    zin[idx] = 3.f * yc + wfea[c] * we + dog;
}

// ---------------------------------------------------------------------------
// Kernel 4: 1x1 mixer as fp32 WMMA GEMM + skip residual, with TDM-staged B.
//
// One workgroup = 16 waves (512 thr) covers ALL 256 output channels (one
// 16-row M-tile per wave) over a 64-wide N-slice of one batch image.
// The B panel zin[b][0:256][n0:n0+64] (256x64 f32 = 64 KB) is DMA'd into LDS
// once by the Tensor Data Mover (D# per ISA §8.3/8.4: 2-D tile, data_size=4B,
// tile_dim0=64, tile_dim1=256, tensor_dim0_stride=9216), so every zin element
// crosses the fabric exactly once. Inner loop: 2 L2 weight loads + 8 ds_load
// + 4 V_WMMA_F32_16X16X4_F32 per K-step.
// ---------------------------------------------------------------------------
__global__ void __launch_bounds__(512) mixer_wmma_kernel(
        const float* __restrict__ zin, const float* __restrict__ Wm,
        const float* __restrict__ skip, float* __restrict__ out) {
    __shared__ float btile[NC * 64];          // 64 KB B panel (K x N)

    int b  = blockIdx.x / 144;                // batch
    int n0 = (blockIdx.x % 144) * 64;         // N-slice start
    int lane = (int)(threadIdx.x & 31);
    int m0 = (int)(threadIdx.x >> 5) * 16;    // this wave's 16 output rows
    int ml = lane & 15;                       // row (A) / col (B) within tile
    int kh = (lane >> 4) << 1;                // K sub-base for half-wave: 0|2

    const float* zpanel = zin + (size_t)b * NC * NHW + n0;

#if __has_builtin(__builtin_amdgcn_tensor_load_to_lds)
    if (threadIdx.x < 32) {
        unsigned long long ga = (unsigned long long)(const void*)zpanel;
        unsigned lds_off = (unsigned)(size_t)(void*)btile;  // low 32b = LDS offset
        // D# group0: count=1 | lds_addr | global_addr[56:0] | type=2
        u32x4 g0;
        g0[0] = 1u;
        g0[1] = lds_off;
        g0[2] = (unsigned)ga;
        g0[3] = ((unsigned)(ga >> 32) & 0x1FFFFFFu) | (2u << 30);
        // D# group1: data_size=4B; tensor_dim0=9216, dim1=256;
        //            tile_dim0=64, tile_dim1=256; dim0_stride=9216
        i32x8 g1;
        g1[0] = (int)(2u << 16);                         // data_size
        g1[1] = (int)((unsigned)(NHW & 0xFFFF) << 16);   // dim0[15:0]
        g1[2] = (int)((NHW >> 16) | ((NC & 0xFFFF) << 16));
        g1[3] = (int)((NC >> 16) | (64 << 16));          // tile_dim0=64
        g1[4] = 256;                                     // tile_dim1=256
        g1[5] = NHW;                                     // dim0_stride
        g1[6] = 0;
        g1[7] = 0;
        i32x4 g2 = {0, 0, 0, 0};
        i32x4 g3 = {0, 0, 0, 0};
#if __clang_major__ >= 23
        i32x8 g4 = {0, 0, 0, 0, 0, 0, 0, 0};
        __builtin_amdgcn_tensor_load_to_lds(g0, g1, g2, g3, g4, 0);
#else
        __builtin_amdgcn_tensor_load_to_lds(g0, g1, g2, g3, 0);
#endif
        __builtin_amdgcn_s_wait_tensorcnt(0);
    }
    __syncthreads();
#else
    // Fallback: cooperative copy global -> LDS
    for (int i = (int)threadIdx.x; i < NC * 64; i += 512) {
        btile[i] = zpanel[(size_t)(i >> 6) * NHW + (i & 63)];
    }
    __syncthreads();
#endif

    const float* wrow = Wm + (size_t)(m0 + ml) * NC;     // weight row (L2-hot)
    __builtin_prefetch(wrow, 0, 0);                      // global_prefetch_b8

    v8f acc[4] = {};
#pragma unroll 4
    for (int k = 0; k < NC; k += 4) {
        v2f a;
        a.x = wrow[k + kh];
        a.y = wrow[k + kh + 1];
        const float* bp = &btile[(k + kh) * 64 + ml];
#pragma unroll
        for (int j = 0; j < 4; ++j) {
            v2f bb;
            bb.x = bp[j * 16];          // row k+kh
            bb.y = bp[64 + j * 16];     // row k+kh+1
            acc[j] = __builtin_amdgcn_wmma_f32_16x16x4_f32(
                false, a, false, bb, (short)0, acc[j], false, false);
        }
    }

    // Epilogue: D layout -> rows m0 + v (+8 for upper half-wave), col n0+ml
    int row0 = m0 + ((lane >> 4) << 3);
    int col0 = n0 + ml;
#pragma unroll
    for (int v = 0; v < 8; ++v) {
        size_t o = ((size_t)b * NC + row0 + v) * NHW + col0;
#pragma unroll
        for (int j = 0; j < 4; ++j) {
            out[o + j * 16] = acc[j][v] + skip[o + j * 16];
        }
    }
}

// ---------------------------------------------------------------------------
// Launch.  inputs: 0=skip 1=dec 2=w_fea 3=sigma1 4=sigma2 5=mixer_w (fp32)
// ---------------------------------------------------------------------------
extern "C" void kernel_launch(void* const* d_in, const int* in_sizes, int n_in,
                              void* d_out, int out_size, void* d_ws, size_t ws_size,
                              hipStream_t stream) {
    const float* skip = (const float*)d_in[0];
    const float* dec  = (const float*)d_in[1];
    const float* wfea = (const float*)d_in[2];
    const float* sig1 = (const float*)d_in[3];
    const float* sig2 = (const float*)d_in[4];
    const float* mixw = (const float*)d_in[5];
    float* out = (float*)d_out;

    // Workspace layout (floats): d24 | d48 | d72 | kdiff | zin  (~283 MB)
    float* ws  = (float*)d_ws;
    float* d24 = ws;
    float* d48 = d24 + (size_t)NB * NC * 24 * 24;
    float* d72 = d48 + (size_t)NB * NC * 48 * 48;
    float* kd  = d72 + (size_t)NB * NC * 72 * 72;
    float* zin = kd + (size_t)NC * 9;

    kdiff_kernel<<<1, 256, 0, stream>>>(sig1, sig2, kd);

    down_kernel<<<(NB * NC * 24 * 24 + 255) / 256, 256, 0, stream>>>(dec, skip, d24, 24);
    down_kernel<<<(NB * NC * 48 * 48 + 255) / 256, 256, 0, stream>>>(dec, skip, d48, 48);
    down_kernel<<<(NB * NC * 72 * 72 + 255) / 256, 256, 0, stream>>>(dec, skip, d72, 72);

    fuse_kernel<<<(NB * NC * NHW) / 256, 256, 0, stream>>>(
        dec, skip, d24, d48, d72, kd, wfea, zin);

    // 16 batches x 144 N-slices; each block = 16 waves covering all 256 M rows
    mixer_wmma_kernel<<<NB * 144, 512, 0, stream>>>(zin, mixw, skip, out);
}